// CausalSelfAttention_48696339202404
// MI455X (gfx1250) — compile-verified
//
#include <hip/hip_runtime.h>

// ---------------------------------------------------------------------------
// CausalSelfAttention for MI455X (gfx1250), full-fp32 path via
// V_WMMA_F32_16X16X4_F32 (wave32 WMMA), with CDNA5 async global->LDS copies
// (ASYNCcnt) for double-buffered tile streaming.
// Shapes fixed: B=1, T=4096, dim=1024, H=8, D=128, scale=0.12.
// ---------------------------------------------------------------------------

typedef __attribute__((ext_vector_type(2))) float v2f;
typedef __attribute__((ext_vector_type(4))) float f4v;
typedef __attribute__((ext_vector_type(8))) float v8f;
typedef __attribute__((__vector_size__(4 * sizeof(int)))) int i32x4;

#define T_SEQ 4096
#define DIM   1024
#define NH    8
#define HD    128
#define ATTN_SCALE 0.12f

#if defined(__HIP_DEVICE_COMPILE__) && __has_builtin(__builtin_amdgcn_global_load_async_to_lds_b128)
#define ASYNC_LDS 1
#else
#define ASYNC_LDS 0
#endif

__device__ __forceinline__ v8f wmma_f32(v2f a, v2f b, v8f c) {
  // D = A(16x4,f32) * B(4x16,f32) + C(16x16,f32)
  return __builtin_amdgcn_wmma_f32_16x16x4_f32(false, a, false, b, (short)0, c,
                                               false, false);
}

__device__ __forceinline__ v8f vzero8() {
  v8f z;
#pragma unroll
  for (int i = 0; i < 8; ++i) z[i] = 0.0f;
  return z;
}

// 16-byte async copy global -> LDS (GLOBAL_LOAD_ASYNC_TO_LDS_B128, ASYNCcnt).
__device__ __forceinline__ void async_ld16(void* lds, const void* g) {
#if ASYNC_LDS
  __builtin_amdgcn_global_load_async_to_lds_b128(
      (__attribute__((address_space(1))) i32x4*)g,
      (__attribute__((address_space(3))) i32x4*)lds, 0, 0);
#else
  *(f4v*)lds = *(const f4v*)g;
#endif
}

__device__ __forceinline__ void async_wait0() {
#if ASYNC_LDS && __has_builtin(__builtin_amdgcn_s_wait_asynccnt)
  __builtin_amdgcn_s_wait_asynccnt(0);
#endif
}

// ---------------------------------------------------------------------------
// Generic C[M,N] = A[M,K] * B[N,K]^T  (row-major A, weights stored [N,K]).
// Block: 128 threads (4 waves). Block tile 64x64, BK=32, double-buffered LDS
// fed by async global->LDS copies; next K-slab streams in while WMMA runs.
// Wave w -> 32x32 subtile as 2x2 fragments of 16x16, K stepped by 4.
// LDS pitch 36 floats: rows 16B aligned, fragment reads bank-conflict free.
// ---------------------------------------------------------------------------
__global__ __launch_bounds__(128) void gemm_nt(const float* __restrict__ A,
                                               const float* __restrict__ B,
                                               float* __restrict__ C,
                                               int M, int N, int K) {
  __shared__ float As[2][64][36];
  __shared__ float Bs[2][64][36];

  const int tid  = threadIdx.x;
  const int lane = tid & 31;
  const int wave = tid >> 5;
  const int half = lane >> 4;   // 0: K-lo half, 1: K-hi half (A/B frags)
  const int l16  = lane & 15;
  const int m0   = blockIdx.y * 64;
  const int n0   = blockIdx.x * 64;
  const int wm   = (wave >> 1) * 32;
  const int wn   = (wave & 1) * 32;

  v8f acc[2][2];
#pragma unroll
  for (int mi = 0; mi < 2; ++mi)
#pragma unroll
    for (int ni = 0; ni < 2; ++ni) acc[mi][ni] = vzero8();

  const int nt = K / 32;

  // Stage slab 0 into buffer 0 (8 async b128 per thread).
#pragma unroll
  for (int it = 0; it < 4; ++it) {
    int idx = tid + it * 128;               // 0..511 float4 slots
    int r = idx >> 3, c4 = (idx & 7) << 2;  // row 0..63, col 0..28
    async_ld16(&As[0][r][c4], A + (size_t)(m0 + r) * K + c4);
    async_ld16(&Bs[0][r][c4], B + (size_t)(n0 + r) * K + c4);
  }
  async_wait0();
  __syncthreads();

  for (int t = 0; t < nt; ++t) {
    const int cb = t & 1;
    // Prefetch next slab into the other buffer while computing this one.
    if (t + 1 < nt) {
      const int kk = (t + 1) * 32;
      const int nb = (t + 1) & 1;
#pragma unroll
      for (int it = 0; it < 4; ++it) {
        int idx = tid + it * 128;
        int r = idx >> 3, c4 = (idx & 7) << 2;
        async_ld16(&As[nb][r][c4], A + (size_t)(m0 + r) * K + kk + c4);
        async_ld16(&Bs[nb][r][c4], B + (size_t)(n0 + r) * K + kk + c4);
      }
    }

#pragma unroll
    for (int k = 0; k < 32; k += 4) {
      const int ko = k + half * 2;  // lanes 16-31 carry K=+2,+3 (ISA A/B layout)
      v2f a[2], b[2];
#pragma unroll
      for (int mi = 0; mi < 2; ++mi) {
        const float* p = &As[cb][wm + mi * 16 + l16][ko];
        a[mi] = (v2f){p[0], p[1]};
      }
#pragma unroll
      for (int ni = 0; ni < 2; ++ni) {
        const float* p = &Bs[cb][wn + ni * 16 + l16][ko];
        b[ni] = (v2f){p[0], p[1]};
      }
#pragma unroll
      for (int mi = 0; mi < 2; ++mi)
#pragma unroll
        for (int ni = 0; ni < 2; ++ni)
          acc[mi][ni] = wmma_f32(a[mi], b[ni], acc[mi][ni]);
    }

    async_wait0();     // next slab resident
    __syncthreads();   // all waves done with current buffer before reuse
  }

  // C/D layout: VGPR i -> row i (lanes 0-15) / row i+8 (lanes 16-31), col = l16
#pragma unroll
  for (int mi = 0; mi < 2; ++mi)
#pragma unroll
    for (int ni = 0; ni < 2; ++ni) {
      int rbase = m0 + wm + mi * 16 + half * 8;
      int col   = n0 + wn + ni * 16 + l16;
#pragma unroll
      for (int i = 0; i < 8; ++i)
        C[(size_t)(rbase + i) * N + col] = acc[mi][ni][i];
    }
}

// ---------------------------------------------------------------------------
// Post-QKV: per (t,h) row of 128 -> RMS-norm(q,k), RoPE(q,k), v-mix.
// One wave per (t,h); lane l owns d = l, l+32, l+64, l+96.
// RoPE pairs (d, d+64): pair j=l rotates with freq (1/1024)^(l/31), pair
// j=l+32 has freq 0 (identity). Outputs head-major [H][T][D].
// ---------------------------------------------------------------------------
__global__ __launch_bounds__(128) void qkv_post(const float* __restrict__ qkv,
                                                const float* __restrict__ ve,
                                                const float* __restrict__ lambdas,
                                                float* __restrict__ qb,
                                                float* __restrict__ kb,
                                                float* __restrict__ vb) {
  const int gw   = blockIdx.x * 4 + (threadIdx.x >> 5);  // 0 .. T*H-1
  const int lane = threadIdx.x & 31;
  const int t = gw >> 3;
  const int h = gw & 7;
  const float l0 = lambdas[0], l1 = lambdas[1];

  const size_t base = (size_t)t * (3 * NH * HD) + (size_t)h * HD;
  float qv[4], kv[4];
#pragma unroll
  for (int i = 0; i < 4; ++i) {
    int d = lane + i * 32;
    qv[i] = qkv[base + d];
    kv[i] = qkv[base + NH * HD + d];
  }

  float sq = 0.f, sk = 0.f;
#pragma unroll
  for (int i = 0; i < 4; ++i) { sq += qv[i] * qv[i]; sk += kv[i] * kv[i]; }
#pragma unroll
  for (int m = 1; m < 32; m <<= 1) {
    sq += __shfl_xor(sq, m, 32);
    sk += __shfl_xor(sk, m, 32);
  }
  const float eps = 1.1920929e-07f;  // finfo(f32).eps
  const float rq = rsqrtf(sq * (1.0f / 128.0f) + eps);
  const float rk = rsqrtf(sk * (1.0f / 128.0f) + eps);
#pragma unroll
  for (int i = 0; i < 4; ++i) { qv[i] *= rq; kv[i] *= rk; }

  // RoPE on pair (elem0, elem2); pair (elem1, elem3) has zero frequency.
  const float freq  = exp2f(-10.0f * (float)lane / 31.0f);
  const float theta = (float)t * freq;
  const float c = cosf(theta), s = sinf(theta);
  {
    float a = qv[0], b2 = qv[2];
    qv[0] = a * c + b2 * s;
    qv[2] = b2 * c - a * s;
    a = kv[0]; b2 = kv[2];
    kv[0] = a * c + b2 * s;
    kv[2] = b2 * c - a * s;
  }

  const size_t ob = ((size_t)h * T_SEQ + t) * HD;
#pragma unroll
  for (int i = 0; i < 4; ++i) {
    int d = lane + i * 32;
    qb[ob + d] = qv[i];
    kb[ob + d] = kv[i];
    float vv = qkv[base + 2 * NH * HD + d];
    vb[ob + d] = l0 * vv + l1 * ve[(size_t)t * DIM + h * HD + d];
  }
}

// ---------------------------------------------------------------------------
// Causal flash attention, streaming softmax, f32 WMMA for QK^T and PV.
// Grid: (T/32 query tiles, H heads). Block: 64 threads = 2 waves,
// each wave owns 16 query rows. 32-row K/V tiles double-buffered in dynamic
// LDS (88 KB -- legal on CDNA5's 320 KB/WGP) and streamed with async copies,
// so the next K/V tile loads while the current tile's WMMAs run.
// Dynamic-LDS float layout:
//   Qs [32][132]            @ 0       (4224)
//   Ks [2][32][132]         @ 4224    (8448)
//   Vs [2][32][128]         @ 12672   (8192)
//   Ps [2][16][36]          @ 20864   (1152)   total 22016 f = 88064 B
// ---------------------------------------------------------------------------
#define FLASH_LDS_BYTES (22016 * 4)

__global__ __launch_bounds__(64) void flash_attn(const float* __restrict__ qb,
                                                 const float* __restrict__ kb,
                                                 const float* __restrict__ vb,
                                                 float* __restrict__ yb) {
  extern __shared__ float smem[];
  float (*Qs)[132]      = (float (*)[132])smem;
  float (*Ks)[32][132]  = (float (*)[32][132])(smem + 4224);
  float (*Vs)[32][128]  = (float (*)[32][128])(smem + 4224 + 8448);
  float (*Ps)[16][36]   = (float (*)[16][36])(smem + 4224 + 8448 + 8192);

  const int tid  = threadIdx.x;
  const int lane = tid & 31;
  const int wave = tid >> 5;
  const int half = lane >> 4;
  const int l16  = lane & 15;
  const int qblk = blockIdx.x;          // 0..127
  const int h    = blockIdx.y;
  const int q0   = qblk * 32;
  const size_t hb = (size_t)h * T_SEQ * HD;

  // Stage Q tile (32x128) + K/V tile 0 asynchronously.
#pragma unroll
  for (int it = 0; it < 16; ++it) {
    int idx = tid + it * 64;                 // 0..1023 float4 slots
    int r = idx >> 5, c4 = (idx & 31) << 2;  // row 0..31, col 0..124
    async_ld16(&Qs[r][c4],    qb + hb + (size_t)(q0 + r) * HD + c4);
    async_ld16(&Ks[0][r][c4], kb + hb + (size_t)r * HD + c4);
    async_ld16(&Vs[0][r][c4], vb + hb + (size_t)r * HD + c4);
  }
  async_wait0();
  __syncthreads();

  v8f o[8];
#pragma unroll
  for (int dt = 0; dt < 8; ++dt) o[dt] = vzero8();
  float Mi[8], Li[8];
#pragma unroll
  for (int i = 0; i < 8; ++i) { Mi[i] = -1e30f; Li[i] = 0.0f; }

  for (int kt = 0; kt <= qblk; ++kt) {
    const int cb = kt & 1;

    // Prefetch next K/V tile into the other buffer (overlaps with compute).
    if (kt + 1 <= qblk) {
      const int nb = (kt + 1) & 1;
      const size_t sb = (size_t)(kt + 1) * 32;
#pragma unroll
      for (int it = 0; it < 16; ++it) {
        int idx = tid + it * 64;
        int r = idx >> 5, c4 = (idx & 31) << 2;
        async_ld16(&Ks[nb][r][c4], kb + hb + (sb + r) * HD + c4);
        async_ld16(&Vs[nb][r][c4], vb + hb + (sb + r) * HD + c4);
      }
    }

    const int s0g = kt * 32;

    // S = Q * K^T : rows [wave*16,+16), key cols [0,32) as 2 fragments
    v8f sf[2];
    sf[0] = vzero8();
    sf[1] = vzero8();
#pragma unroll
    for (int cc = 0; cc < HD; cc += 4) {
      const int co = cc + half * 2;
      const float* qp = &Qs[wave * 16 + l16][co];
      v2f a = (v2f){qp[0], qp[1]};
#pragma unroll
      for (int ni = 0; ni < 2; ++ni) {
        const float* kp = &Ks[cb][ni * 16 + l16][co];
        v2f b = (v2f){kp[0], kp[1]};
        sf[ni] = wmma_f32(a, b, sf[ni]);
      }
    }

    // Scale + causal mask (branch-free per lane; kt check is uniform)
    const bool diag = (kt == qblk);
#pragma unroll
    for (int ni = 0; ni < 2; ++ni)
#pragma unroll
      for (int i = 0; i < 8; ++i) {
        float v = sf[ni][i] * ATTN_SCALE;
        if (diag) {
          int qr = q0 + wave * 16 + i + half * 8;
          int kc = s0g + ni * 16 + l16;
          v = (kc > qr) ? -1e30f : v;
        }
        sf[ni][i] = v;
      }

    // Online softmax update (per-row stats replicated across each 16-lane half)
#pragma unroll
    for (int i = 0; i < 8; ++i) {
      float rm = fmaxf(sf[0][i], sf[1][i]);
#pragma unroll
      for (int m = 1; m < 16; m <<= 1) rm = fmaxf(rm, __shfl_xor(rm, m, 32));
      float mnew  = fmaxf(Mi[i], rm);
      float alpha = __expf(Mi[i] - mnew);
      Mi[i] = mnew;
      float p0 = __expf(sf[0][i] - mnew);
      float p1 = __expf(sf[1][i] - mnew);
      sf[0][i] = p0;
      sf[1][i] = p1;
      float rs = p0 + p1;
#pragma unroll
      for (int m = 1; m < 16; m <<= 1) rs += __shfl_xor(rs, m, 32);
      Li[i] = Li[i] * alpha + rs;
#pragma unroll
      for (int dt = 0; dt < 8; ++dt) o[dt][i] *= alpha;
    }

    // C-layout P -> LDS (row-major 16x32 per wave)
#pragma unroll
    for (int i = 0; i < 8; ++i) {
      int row = i + half * 8;
      Ps[wave][row][l16]      = sf[0][i];
      Ps[wave][row][l16 + 16] = sf[1][i];
    }
    asm volatile("" ::: "memory");  // keep DS write->read order (HW is in-order)

    // O += P * V : A = P (16 x 32), B = V chunk (32 x 128) as 8 d-tiles
#pragma unroll
    for (int ss = 0; ss < 32; ss += 4) {
      const int so = ss + half * 2;
      const float* pp = &Ps[wave][l16][so];
      v2f a = (v2f){pp[0], pp[1]};
#pragma unroll
      for (int dt = 0; dt < 8; ++dt) {
        const float* vp = &Vs[cb][so][dt * 16 + l16];
        v2f b = (v2f){vp[0], vp[128]};  // rows so, so+1 (stride 128 floats)
        o[dt] = wmma_f32(a, b, o[dt]);
      }
    }

    async_wait0();     // next K/V tile resident
    __syncthreads();   // all waves done with current buffer before reuse
  }

  // Normalize and write y[T][H*D]
#pragma unroll
  for (int i = 0; i < 8; ++i) {
    float inv = 1.0f / Li[i];
    int row = q0 + wave * 16 + i + half * 8;
#pragma unroll
    for (int dt = 0; dt < 8; ++dt)
      yb[(size_t)row * DIM + h * HD + dt * 16 + l16] = o[dt][i] * inv;
  }
}

// ---------------------------------------------------------------------------
extern "C" void kernel_launch(void* const* d_in, const int* in_sizes, int n_in,
                              void* d_out, int out_size, void* d_ws,
                              size_t ws_size, hipStream_t stream) {
  const float* x       = (const float*)d_in[0];  // [1,4096,1024]
  const float* ve      = (const float*)d_in[1];  // [1,4096,1024]
  const float* qkv_w   = (const float*)d_in[2];  // [3,1024,1024] -> [3072,1024]
  const float* lambdas = (const float*)d_in[3];  // [2]
  const float* c_proj  = (const float*)d_in[4];  // [1024,1024]
  float* out = (float*)d_out;                    // [1,4096,1024]

  float* ws   = (float*)d_ws;
  float* qkv  = ws;                       // 4096*3072 floats (48 MB)
  float* yb   = ws;                       // reuse qkv region after qkv_post
  float* qbuf = ws + (size_t)T_SEQ * 3 * DIM;           // +12,582,912
  float* kbuf = qbuf + (size_t)T_SEQ * DIM;             // +4,194,304
  float* vbuf = kbuf + (size_t)T_SEQ * DIM;             // +4,194,304
  // total ws: 96 MB of f32

  // Allow 88 KB dynamic LDS for the flash kernel (CDNA5: 320 KB per WGP).
  // Host-side attribute set; not a stream op, safe under graph capture.
  static int lds_cap_set = 0;
  if (!lds_cap_set) {
    (void)hipFuncSetAttribute((const void*)flash_attn,
                              hipFuncAttributeMaxDynamicSharedMemorySize,
                              FLASH_LDS_BYTES);
    lds_cap_set = 1;
  }

  // 1) qkv = x @ qkv_w^T   [4096 x 3072]
  gemm_nt<<<dim3(3 * DIM / 64, T_SEQ / 64), 128, 0, stream>>>(
      x, qkv_w, qkv, T_SEQ, 3 * DIM, DIM);

  // 2) rmsnorm + rope + v-mix -> head-major q/k/v
  qkv_post<<<(T_SEQ * NH) / 4, 128, 0, stream>>>(qkv, ve, lambdas, qbuf, kbuf,
                                                 vbuf);

  // 3) causal flash attention -> y [4096 x 1024]
  flash_attn<<<dim3(T_SEQ / 32, NH), 64, FLASH_LDS_BYTES, stream>>>(
      qbuf, kbuf, vbuf, yb);

  // 4) out = y @ c_proj^T   [4096 x 1024]
  gemm_nt<<<dim3(DIM / 64, T_SEQ / 64), 128, 0, stream>>>(
      yb, c_proj, out, T_SEQ, DIM, DIM);
}